// HeteroScalarProteinGNN_HEAT_18013092839587
// MI455X (gfx1250) — compile-verified
//
#include <hip/hip_runtime.h>
#include <hip/hip_bf16.h>

// ---------------- problem constants (match reference) ----------------
#define N_NODES 50000
#define N_EDGES 1600000
#define F_IN    128
#define F_OUT   32
#define N_HEAD  2
#define N_NT    20
#define N_ET    4
#define D_ETE   16
#define D_EAE   16
#define D_ED    16
#define D_ATT   96          // 2*F_OUT + D_ETE + D_EAE
#define SLOPE   0.2f

#define NODE_TILES (N_NODES / 16)   // 3125 (N divisible by 16)
#define EDGE_TILES (N_EDGES / 16)   // 100000 (E divisible by 16)

typedef __attribute__((ext_vector_type(2))) float v2f;
typedef __attribute__((ext_vector_type(8))) float v8f;

__device__ __forceinline__ float lrelu(float v) { return v > 0.f ? v : SLOPE * v; }

// ---------------- init: zero out/denom, amax = -inf ----------------
__global__ void k_init(float* __restrict__ amax, float* __restrict__ denom,
                       float* __restrict__ out) {
    int i = blockIdx.x * blockDim.x + threadIdx.x;
    if (i < N_NODES * N_HEAD) { amax[i] = -3.4e38f; denom[i] = 0.f; }
    if (i < N_NODES * N_HEAD * F_OUT) out[i] = 0.f;
}

// ---------------- K1: hetero linear, fp32 WMMA 16x16x4 ----------------
// One wave per 16-node tile. A (x tile, 16x128) hoisted into 32 v2f frags.
// Loop over the 20 node types, skip absent ones, mask-accumulate D rows.
__global__ __launch_bounds__(256) void k_hetero(
        const float* __restrict__ x, const int* __restrict__ ntypes,
        const float* __restrict__ hetW, const float* __restrict__ hetb,
        float* __restrict__ h) {
    const int wave  = threadIdx.x >> 5;
    const int lane  = threadIdx.x & 31;
    const int tile  = blockIdx.x * 8 + wave;
    if (tile >= NODE_TILES) return;
    const int base  = tile * 16;
    const int row16 = lane & 15;
    const int khalf = lane >> 4;     // lanes 16-31 carry K+2,K+3 of each frag

    // A fragments: a[kk] = x[base+row16][kk*4 + khalf*2 + {0,1}]
    v2f a[32];
    const float* xr = x + (size_t)(base + row16) * F_IN + khalf * 2;
#pragma unroll
    for (int kk = 0; kk < 32; ++kk)
        a[kk] = *(const v2f*)(xr + kk * 4);

    // D-row mapping and node types for the rows this lane stores
    int rowD[8], typ[8];
#pragma unroll
    for (int r = 0; r < 8; ++r) {
        rowD[r] = (lane < 16) ? r : r + 8;
        typ[r]  = ntypes[base + rowD[r]];
    }

    float acc[2][8];
#pragma unroll
    for (int n2 = 0; n2 < 2; ++n2)
#pragma unroll
        for (int r = 0; r < 8; ++r) acc[n2][r] = 0.f;

    for (int t = 0; t < N_NT; ++t) {
        int mine = 0;
#pragma unroll
        for (int r = 0; r < 8; ++r) mine |= (typ[r] == t);
        if (!__any(mine)) continue;                   // type absent in tile
        const float* W = hetW + (size_t)t * F_IN * F_OUT;
#pragma unroll
        for (int n2 = 0; n2 < 2; ++n2) {
            const int ncol = n2 * 16 + row16;
            v8f c = {0.f, 0.f, 0.f, 0.f, 0.f, 0.f, 0.f, 0.f};
#pragma unroll
            for (int kk = 0; kk < 32; ++kk) {
                const int k0 = kk * 4 + khalf * 2;
                v2f b;
                b.x = W[(k0 + 0) * F_OUT + ncol];
                b.y = W[(k0 + 1) * F_OUT + ncol];
                c = __builtin_amdgcn_wmma_f32_16x16x4_f32(
                        false, a[kk], false, b, (short)0, c, false, false);
            }
#pragma unroll
            for (int r = 0; r < 8; ++r)
                if (typ[r] == t) acc[n2][r] += c[r];
        }
    }

    // bias + store h
#pragma unroll
    for (int n2 = 0; n2 < 2; ++n2) {
        const int ncol = n2 * 16 + row16;
#pragma unroll
        for (int r = 0; r < 8; ++r) {
            float v = acc[n2][r] + hetb[typ[r] * F_OUT + ncol];
            h[(size_t)(base + rowD[r]) * F_OUT + ncol] = v;
        }
    }
}

// ---------------- K2: per-edge attention logits + segment max ----------------
__global__ void k_edge_logits(
        const int* __restrict__ ei, const int* __restrict__ etypes,
        const float* __restrict__ eattr, const float* __restrict__ etab,
        const float* __restrict__ eaW, const float* __restrict__ attW,
        const float* __restrict__ h, float* __restrict__ alpha,
        float* __restrict__ amax) {
    const int e = blockIdx.x * blockDim.x + threadIdx.x;
    if (e >= N_EDGES) return;
    const int s  = ei[e];
    const int d  = ei[N_EDGES + e];
    const int et = etypes[e];

    float ete[D_ETE];
#pragma unroll
    for (int j = 0; j < D_ETE; ++j) ete[j] = lrelu(etab[et * D_ETE + j]);

    float er[D_ED];
    const float* eat = eattr + (size_t)e * D_ED;
#pragma unroll
    for (int k = 0; k < D_ED; ++k) er[k] = eat[k];

    float ea[D_EAE];
#pragma unroll
    for (int j = 0; j < D_EAE; ++j) {
        float sum = 0.f;
#pragma unroll
        for (int k = 0; k < D_ED; ++k) sum += er[k] * eaW[k * D_EAE + j];
        ea[j] = lrelu(sum);
    }

    const float* xi = h + (size_t)d * F_OUT;
    const float* xj = h + (size_t)s * F_OUT;
#pragma unroll
    for (int hh = 0; hh < N_HEAD; ++hh) {
        float sum = 0.f;
#pragma unroll
        for (int c = 0; c < F_OUT; ++c) sum += xi[c] * attW[c * N_HEAD + hh];
#pragma unroll
        for (int c = 0; c < F_OUT; ++c) sum += xj[c] * attW[(F_OUT + c) * N_HEAD + hh];
#pragma unroll
        for (int j = 0; j < D_ETE; ++j) sum += ete[j] * attW[(2 * F_OUT + j) * N_HEAD + hh];
#pragma unroll
        for (int j = 0; j < D_EAE; ++j) sum += ea[j] * attW[(2 * F_OUT + D_ETE + j) * N_HEAD + hh];
        const float ar = lrelu(sum);
        alpha[(size_t)e * N_HEAD + hh] = ar;
        atomicMax(&amax[d * N_HEAD + hh], ar);   // native global_atomic_max_num_f32
    }
}

// ---------------- K3: exp(alpha - amax[dst]) + segment sum ----------------
__global__ void k_edge_exp(const int* __restrict__ ei,
                           const float* __restrict__ amax,
                           float* __restrict__ alpha,
                           float* __restrict__ denom) {
    const int e = blockIdx.x * blockDim.x + threadIdx.x;
    if (e >= N_EDGES) return;
    const int d = ei[N_EDGES + e];
#pragma unroll
    for (int hh = 0; hh < N_HEAD; ++hh) {
        const float av = __expf(alpha[(size_t)e * N_HEAD + hh] - amax[d * N_HEAD + hh]);
        alpha[(size_t)e * N_HEAD + hh] = av;
        atomicAdd(&denom[d * N_HEAD + hh], av);
    }
}

// ---------------- K4: message GEMM (fp32 WMMA) + weighted scatter ----------------
// One wave per 16-edge tile. A = [x_j | lrelu(eattr@eaW)] (16x48), B = lin_W (48x32).
__global__ __launch_bounds__(256) void k_aggregate(
        const int* __restrict__ ei, const float* __restrict__ eattr,
        const float* __restrict__ eaW, const float* __restrict__ linW,
        const float* __restrict__ h, const float* __restrict__ alpha,
        const float* __restrict__ denom, float* __restrict__ out) {
    const int wave  = threadIdx.x >> 5;
    const int lane  = threadIdx.x & 31;
    const int tile  = blockIdx.x * 8 + wave;
    if (tile >= EDGE_TILES) return;
    const int ebase = tile * 16;
    const int row16 = lane & 15;
    const int khalf = lane >> 4;

    // recompute ea for this lane's A-row edge (lanes L and L+16 share a row)
    const int eRow   = ebase + row16;
    const int srcRow = ei[eRow];
    float ea[D_EAE];
    {
        float er[D_ED];
        const float* eat = eattr + (size_t)eRow * D_ED;
#pragma unroll
        for (int k = 0; k < D_ED; ++k) er[k] = eat[k];
#pragma unroll
        for (int j = 0; j < D_EAE; ++j) {
            float s = 0.f;
#pragma unroll
            for (int k = 0; k < D_ED; ++k) s += er[k] * eaW[k * D_EAE + j];
            ea[j] = lrelu(s);
        }
    }
    const float* xj = h + (size_t)srcRow * F_OUT;

    // A fragments over K = 48 (cols 0-31 = x_j, cols 32-47 = ea)
    v2f a[12];
#pragma unroll
    for (int kk = 0; kk < 12; ++kk) {
        const int k0 = kk * 4 + khalf * 2;
        const float a0 = (k0     < F_OUT) ? xj[k0]     : ea[k0 - F_OUT];
        const float a1 = (k0 + 1 < F_OUT) ? xj[k0 + 1] : ea[k0 + 1 - F_OUT];
        v2f av; av.x = a0; av.y = a1;
        a[kk] = av;
    }

    float msg[2][8];
#pragma unroll
    for (int n2 = 0; n2 < 2; ++n2) {
        const int ncol = n2 * 16 + row16;
        v8f c = {0.f, 0.f, 0.f, 0.f, 0.f, 0.f, 0.f, 0.f};
#pragma unroll
        for (int kk = 0; kk < 12; ++kk) {
            const int k0 = kk * 4 + khalf * 2;
            v2f b;
            b.x = linW[(k0 + 0) * F_OUT + ncol];
            b.y = linW[(k0 + 1) * F_OUT + ncol];
            c = __builtin_amdgcn_wmma_f32_16x16x4_f32(
                    false, a[kk], false, b, (short)0, c, false, false);
        }
#pragma unroll
        for (int r = 0; r < 8; ++r) msg[n2][r] = c[r];
    }

    // attention-scaled scatter: out[dst, hh*32 + col] += msg * alpha/denom
#pragma unroll
    for (int r = 0; r < 8; ++r) {
        const int rowD = (lane < 16) ? r : r + 8;
        const int eOut = ebase + rowD;
        const int dN   = ei[N_EDGES + eOut];
        const float a0 = alpha[(size_t)eOut * N_HEAD + 0] / denom[dN * N_HEAD + 0];
        const float a1 = alpha[(size_t)eOut * N_HEAD + 1] / denom[dN * N_HEAD + 1];
#pragma unroll
        for (int n2 = 0; n2 < 2; ++n2) {
            const int ncol = n2 * 16 + row16;
            const float v  = msg[n2][r];
            atomicAdd(&out[(size_t)dN * (N_HEAD * F_OUT) + ncol],         v * a0);
            atomicAdd(&out[(size_t)dN * (N_HEAD * F_OUT) + F_OUT + ncol], v * a1);
        }
    }
}

// ---------------- K5: final relu ----------------
__global__ void k_relu(float* __restrict__ out, int n) {
    const int i = blockIdx.x * blockDim.x + threadIdx.x;
    if (i < n) out[i] = fmaxf(out[i], 0.f);
}

// ---------------- launch ----------------
extern "C" void kernel_launch(void* const* d_in, const int* in_sizes, int n_in,
                              void* d_out, int out_size, void* d_ws, size_t ws_size,
                              hipStream_t stream) {
    const float* x      = (const float*)d_in[0];
    const int*   ei     = (const int*)  d_in[1];
    const int*   ntypes = (const int*)  d_in[2];
    const int*   etypes = (const int*)  d_in[3];
    const float* eattr  = (const float*)d_in[4];
    const float* hetW   = (const float*)d_in[5];
    const float* hetb   = (const float*)d_in[6];
    const float* etab   = (const float*)d_in[7];
    const float* eaW    = (const float*)d_in[8];
    const float* attW   = (const float*)d_in[9];
    const float* linW   = (const float*)d_in[10];
    float* out = (float*)d_out;

    // workspace layout (floats): h[N*32] | amax[N*2] | denom[N*2] | alpha[E*2]  (~20 MB)
    float* h     = (float*)d_ws;
    float* amax  = h     + (size_t)N_NODES * F_OUT;
    float* denom = amax  + (size_t)N_NODES * N_HEAD;
    float* alpha = denom + (size_t)N_NODES * N_HEAD;

    const int outElems = N_NODES * N_HEAD * F_OUT;

    k_init<<<(outElems + 255) / 256, 256, 0, stream>>>(amax, denom, out);
    k_hetero<<<(NODE_TILES + 7) / 8, 256, 0, stream>>>(x, ntypes, hetW, hetb, h);
    k_edge_logits<<<(N_EDGES + 255) / 256, 256, 0, stream>>>(
        ei, etypes, eattr, etab, eaW, attW, h, alpha, amax);
    k_edge_exp<<<(N_EDGES + 255) / 256, 256, 0, stream>>>(ei, amax, alpha, denom);
    k_aggregate<<<(EDGE_TILES + 7) / 8, 256, 0, stream>>>(
        ei, eattr, eaW, linW, h, alpha, denom, out);
    k_relu<<<(outElems + 255) / 256, 256, 0, stream>>>(out, outElems);
}